// FastSamPredict_71536975282759
// MI455X (gfx1250) — compile-verified
//
#include <hip/hip_runtime.h>

// FastSAM bbox-from-mask reduction for MI455X (gfx1250).
//
// Roofline: 512 MB of mask data @ 23.3 TB/s HBM => ~22 us floor; the kernel is
// purely bandwidth-bound (no matrix structure => WMMA is not applicable).
// We use the CDNA5 async global->LDS pipeline (GLOBAL_LOAD_ASYNC_TO_LDS_B128 +
// ASYNCcnt) to keep 4 row-chunks per wave in flight without VGPR pressure,
// and bitwise-OR presence accumulation (masks are exactly 0.0f / 1.0f).

namespace {
constexpr int kN = 128;
constexpr int kH = 1024;
constexpr int kW = 1024;
constexpr int kThreads = 256;                    // 8 wave32 per block
constexpr int kRowsPerBlock = 64;
constexpr int kBlocksPerMask = kH / kRowsPerBlock; // 16
constexpr int kRowBytes = kW * 4;                // 4096 B per row
constexpr int kSlots = 8;                        // LDS ring: 8 * 4 KB = 32 KB
constexpr int kDepth = 4;                        // async issue-ahead distance
}

__global__ void bbox_init_ws_kernel(int* __restrict__ ws) {
  int i = blockIdx.x * blockDim.x + threadIdx.x;
  if (i < kN) {
    // [xmin, ymin, xmax, ymax] sentinels matching the reference's where() defaults.
    reinterpret_cast<int4*>(ws)[i] = make_int4(kW, kH, -1, -1);
  }
}

__global__ __launch_bounds__(kThreads)
void bbox_reduce_kernel(const float* __restrict__ masks, int* __restrict__ ws) {
  __shared__ __align__(16) unsigned char stage[kSlots * kRowBytes]; // 32 KB ring

  const int t    = threadIdx.x;          // owns columns 4t..4t+3
  const int n    = blockIdx.y;           // mask index
  const int row0 = blockIdx.x * kRowsPerBlock;

  const unsigned long long gbase =
      (unsigned long long)(const unsigned char*)masks +
      ((unsigned long long)n * kH + (unsigned long long)row0) *
          (unsigned long long)kRowBytes +
      (unsigned long long)(t * 16);

  // Low 32 bits of the flat address of a __shared__ object are the LDS byte
  // offset (flat LDS aperture keeps the offset in addr[31:0]).
  const unsigned int lbase = (unsigned int)(unsigned long long)(&stage[t * 16]);

  // ---- prologue: launch kDepth rows of async global->LDS copies (per wave,
  // each instruction moves 32 lanes x 16 B = 512 B; per block = one full row).
  #pragma unroll
  for (int s = 0; s < kDepth; ++s) {
    unsigned long long ga = gbase + (unsigned long long)s * kRowBytes;
    unsigned int       la = lbase + (unsigned int)(s % kSlots) * kRowBytes;
    asm volatile("global_load_async_to_lds_b128 %0, %1, off"
                 :: "v"(la), "v"(ga) : "memory");
  }

  unsigned int a0 = 0, a1 = 0, a2 = 0, a3 = 0; // column presence (bitwise OR)
  int ymin = kH, ymax = -1;

  // ---- steady state: wait oldest async, read own 16 B back from LDS,
  // refill the slot 4 ahead (slot being written was last read 4 rows ago).
  for (int r = 0; r < kRowsPerBlock - kDepth; ++r) {
    asm volatile("s_wait_asynccnt 3" ::: "memory"); // oldest of <=4 done
    const uint4 u = *reinterpret_cast<const uint4*>(
        &stage[(r % kSlots) * kRowBytes + t * 16]);
    {
      unsigned long long ga = gbase + (unsigned long long)(r + kDepth) * kRowBytes;
      unsigned int       la = lbase + (unsigned int)((r + kDepth) % kSlots) * kRowBytes;
      asm volatile("global_load_async_to_lds_b128 %0, %1, off"
                   :: "v"(la), "v"(ga) : "memory");
    }
    a0 |= u.x; a1 |= u.y; a2 |= u.z; a3 |= u.w;
    const unsigned int any = u.x | u.y | u.z | u.w; // nonzero iff some 1.0f
    const int y = row0 + r;
    if (any) { ymax = y; ymin = min(ymin, y); }     // rows ascend -> ymax = y
  }

  // ---- tail: drain remaining asyncs, process last kDepth rows.
  asm volatile("s_wait_asynccnt 0" ::: "memory");
  #pragma unroll
  for (int r = kRowsPerBlock - kDepth; r < kRowsPerBlock; ++r) {
    const uint4 u = *reinterpret_cast<const uint4*>(
        &stage[(r % kSlots) * kRowBytes + t * 16]);
    a0 |= u.x; a1 |= u.y; a2 |= u.z; a3 |= u.w;
    const unsigned int any = u.x | u.y | u.z | u.w;
    const int y = row0 + r;
    if (any) { ymax = y; ymin = min(ymin, y); }
  }

  // ---- per-thread x extremes over its 4 private columns.
  const int c0 = t * 4;
  int xmin = kW, xmax = -1;
  if (a0) { xmin = min(xmin, c0 + 0); xmax = max(xmax, c0 + 0); }
  if (a1) { xmin = min(xmin, c0 + 1); xmax = max(xmax, c0 + 1); }
  if (a2) { xmin = min(xmin, c0 + 2); xmax = max(xmax, c0 + 2); }
  if (a3) { xmin = min(xmin, c0 + 3); xmax = max(xmax, c0 + 3); }

  // ---- wave32 butterfly reduction (all lanes converge).
  #pragma unroll
  for (int m = 16; m >= 1; m >>= 1) {
    xmin = min(xmin, __shfl_xor(xmin, m, 32));
    xmax = max(xmax, __shfl_xor(xmax, m, 32));
    ymin = min(ymin, __shfl_xor(ymin, m, 32));
    ymax = max(ymax, __shfl_xor(ymax, m, 32));
  }

  if ((t & 31) == 0) {
    int* w4 = ws + n * 4;
    atomicMin(w4 + 0, xmin);
    atomicMin(w4 + 1, ymin);
    atomicMax(w4 + 2, xmax);
    atomicMax(w4 + 3, ymax);
  }
}

__global__ void bbox_finalize_kernel(const int* __restrict__ ws,
                                     float* __restrict__ out) {
  int i = blockIdx.x * blockDim.x + threadIdx.x;
  if (i < kN * 4) out[i] = (float)ws[i];   // [n][x1,y1,x2,y2] -> f32
}

extern "C" void kernel_launch(void* const* d_in, const int* in_sizes, int n_in,
                              void* d_out, int out_size, void* d_ws, size_t ws_size,
                              hipStream_t stream) {
  (void)in_sizes; (void)n_in; (void)out_size; (void)ws_size;
  const float* masks = (const float*)d_in[0];
  int*   ws  = (int*)d_ws;     // kN*4 ints of scratch (re-initialized each call)
  float* out = (float*)d_out;

  bbox_init_ws_kernel<<<dim3(1), dim3(kN), 0, stream>>>(ws);

  dim3 grid(kBlocksPerMask, kN);
  bbox_reduce_kernel<<<grid, dim3(kThreads), 0, stream>>>(masks, ws);

  bbox_finalize_kernel<<<dim3((kN * 4 + 255) / 256), dim3(256), 0, stream>>>(ws, out);
}